// NGQS_RES_FNN_56049323213800
// MI455X (gfx1250) — compile-verified
//
#include <hip/hip_runtime.h>
#include <hip/hip_bf16.h>

// ---------------------------------------------------------------------------
// 22-qubit state-vector circuit, restructured for MI455X (gfx1250):
//  * CNOT ladder == binary->Gray permutation (i ^ (i>>1)); folded into
//    addressing, zero extra passes.
//  * Each RY layer = tensor product of 2x2 rotations; applied 4 qubits at a
//    time as 16x16 Kronecker matrices via V_WMMA_F32_16X16X4_F32 on LDS tiles.
//  * Normalization folded into final probability (1/sum of squares).
//  * LDS uses an additive skew  phys(a) = a + (a>>4)  so the stride-16
//    access pattern of the g=0 stage is conflict-free (stride 17 over 64
//    banks) while stride-1 stages stay conflict-free and all offsets still
//    constant-fold (g is a template parameter).
// Memory-bound: ~0.47 GB total traffic -> ~20us at 23.3 TB/s.
// Workspace (floats): [A.re | A.im | B.re | B.im | invn2 (+pad) | partials]
// ---------------------------------------------------------------------------

#define NQ    22
#define DIM   (1u << NQ)        // 4,194,304 amplitudes
#define LOWB  14                // bits per "low" LDS pass tile
#define TILE  (1u << LOWB)      // 16384 amps per plane
#define PTILE (TILE + TILE / 16)// 17408 floats: skewed plane (68KB) x2 = 136KB
#define CONT  6                 // contiguous low bits kept in the high-pass tile
#define NTHREADS 256            // 8 wave32 per workgroup
#define PER_THREAD (TILE / NTHREADS)   // 64

typedef float v2f __attribute__((ext_vector_type(2)));
typedef float v8f __attribute__((ext_vector_type(8)));

__device__ __forceinline__ int phys(int a) { return a + (a >> 4); }

// ---------------------------------------------------------------------------
// Build the 16x16 Kronecker matrix  M = U3 (x) U2 (x) U1 (x) U0  in LDS.
// Group bit t corresponds to amplitude bit (gbit0 + t) of the 22-bit index.
// activeMask selects which t get a real RY rotation (others: identity).
// ---------------------------------------------------------------------------
__device__ __forceinline__ void build_M(float (*Msh)[17], float* rc, float* rs,
                                        const float* cp, int layer, int gbit0,
                                        int activeMask, int tid)
{
    if (tid < 4) {
        int t = tid;
        if (activeMask & (1 << t)) {
            int b = gbit0 + t;          // bit position in flat index
            int q = 21 - b;             // reference qubit index (axis q)
            float th = cp[layer * 22 + q] * 0.5f;
            rc[t] = cosf(th);
            rs[t] = sinf(th);
        } else {
            rc[t] = 1.0f;
            rs[t] = 0.0f;
        }
    }
    __syncthreads();
    int i = tid >> 4, j = tid & 15;     // 256 threads -> one element each
    float m = 1.0f;
#pragma unroll
    for (int t = 0; t < 4; ++t) {
        float c = rc[t], s = rs[t];
        int ib = (i >> t) & 1, jb = (j >> t) & 1;
        // RY: [[c,-s],[s,c]]
        m *= ib ? (jb ? c : s) : (jb ? -s : c);
    }
    Msh[i][j] = m;
    __syncthreads();
}

// ---------------------------------------------------------------------------
// Apply M (16x16) to all 16-amplitude subcubes at group bits [G, G+4) of the
// 14-bit local index, for both real and imag LDS planes.
// D = M x S via 4 accumulating V_WMMA_F32_16X16X4_F32 per 16-column tile.
// Columns (= the 10 non-group bits, x2 planes) are partitioned across the
// 8 waves, so no intra-stage races.
//
// ISA operand layouts (CDNA5 ISA 7.12.2):
//  A 16x4 : lane m%16 holds row m;  VGPR0 = K {0 | 2}, VGPR1 = K {1 | 3}
//  B 4x16 : lane n%16 holds col n;  same K striping as A
//  D 16x16: lane n%16 holds col n;  VGPR v = row v + 8*(lane/16)
//
// phys() is additive over our (base, k<<G) splits because the base's low
// nibble and the offset's low nibble are never both nonzero.
// ---------------------------------------------------------------------------
template <int G>
__device__ __forceinline__ void apply_group(float* sre, float* sim,
                                            float (*Msh)[17], int tid)
{
    int lane = tid & 31;
    int wave = tid >> 5;
    int n16  = lane & 15;
    int half = lane >> 4;               // 0: K+0/1, 1: K+2/3

    for (int tile = wave; tile < 128; tile += 8) {
        int cc   = tile * 16 + n16;     // 0..2047 (bit 10 selects plane)
        float* P = (cc >> 10) ? sim : sre;
        int col  = cc & 1023;
        // scatter the 10 column bits around the 4 group bits, then skew
        int base = ((col >> G) << (G + 4)) | (col & ((1 << G) - 1));
        int pb   = phys(base);

        v8f acc = {};
#pragma unroll
        for (int k = 0; k < 16; k += 4) {
            int ka = k + half * 2;
            v2f A, B;
            A.x = Msh[n16][ka];
            A.y = Msh[n16][ka + 1];
            B.x = P[pb + phys(ka << G)];
            B.y = P[pb + phys((ka + 1) << G)];
            acc = __builtin_amdgcn_wmma_f32_16x16x4_f32(
                false, A, false, B, (short)0, acc, false, false);
        }
#pragma unroll
        for (int v = 0; v < 8; ++v)
            P[pb + phys((v + half * 8) << G)] = acc[v];
    }
}

// ---------------------------------------------------------------------------
// K1: tiny MLP -> y[3]; state[i] = (w3[i].y + b3[i]) + j(w3[i+DIM].y + b3[i+DIM])
//     plus per-block partial sum of squares (for the norm).
// ---------------------------------------------------------------------------
__global__ void k_build(const float* __restrict__ alpha,
                        const float* __restrict__ iawp,
                        const float* __restrict__ w1, const float* __restrict__ b1,
                        const float* __restrict__ w2, const float* __restrict__ b2,
                        const float* __restrict__ w3, const float* __restrict__ b3,
                        float* __restrict__ re, float* __restrict__ im,
                        float* __restrict__ partials)
{
    __shared__ float ys[3];
    __shared__ float red[NTHREADS];
    int tid = threadIdx.x;

    if (tid == 0) {
        float iaw = iawp[0];
        float a0 = alpha[0], a1 = alpha[1], a2 = alpha[2];
        float t0 = tanhf(w1[0]*a0 + w1[1]*a1 + w1[2]*a2 + b1[0]) + a0 * iaw;
        float t1 = tanhf(w1[3]*a0 + w1[4]*a1 + w1[5]*a2 + b1[1]) + a1 * iaw;
        float t2 = tanhf(w1[6]*a0 + w1[7]*a1 + w1[8]*a2 + b1[2]) + a2 * iaw;
        ys[0] = tanhf(w2[0]*t0 + w2[1]*t1 + w2[2]*t2 + b2[0]) + t0 * iaw;
        ys[1] = tanhf(w2[3]*t0 + w2[4]*t1 + w2[5]*t2 + b2[1]) + t1 * iaw;
        ys[2] = tanhf(w2[6]*t0 + w2[7]*t1 + w2[8]*t2 + b2[2]) + t2 * iaw;
    }
    __syncthreads();
    float y0 = ys[0], y1 = ys[1], y2 = ys[2];

    float ss = 0.0f;
#pragma unroll
    for (int k = 0; k < 8; ++k) {
        unsigned i = (unsigned)blockIdx.x * 2048u + (unsigned)k * 256u + tid;
        size_t r0 = (size_t)i * 3u;
        size_t r1 = ((size_t)i + DIM) * 3u;
        float r = w3[r0] * y0 + w3[r0 + 1] * y1 + w3[r0 + 2] * y2 + b3[i];
        float m = w3[r1] * y0 + w3[r1 + 1] * y1 + w3[r1 + 2] * y2 + b3[(size_t)i + DIM];
        re[i] = r;
        im[i] = m;
        ss += r * r + m * m;
    }
    red[tid] = ss;
    __syncthreads();
#pragma unroll
    for (int s = NTHREADS / 2; s > 0; s >>= 1) {
        if (tid < s) red[tid] += red[tid + s];
        __syncthreads();
    }
    if (tid == 0) partials[blockIdx.x] = red[0];
}

// K2: reduce 2048 partials -> 1 / sum(|amp|^2)
__global__ void k_norm(const float* __restrict__ partials, float* __restrict__ invn2)
{
    __shared__ float red[NTHREADS];
    int tid = threadIdx.x;
    float s = 0.0f;
    for (int i = tid; i < 2048; i += NTHREADS) s += partials[i];
    red[tid] = s;
    __syncthreads();
#pragma unroll
    for (int k = NTHREADS / 2; k > 0; k >>= 1) {
        if (tid < k) red[tid] += red[tid + k];
        __syncthreads();
    }
    if (tid == 0) invn2[0] = 1.0f / red[0];
}

// ---------------------------------------------------------------------------
// K3/K5: low pass. One block owns 2^14 consecutive indices (bits 0..13),
// applies rotations on bits 0..13 (qubits 8..21) via WMMA stages
// g = 0, 4, 8 and a partial group at g = 10 (identity on bits 10-11).
// gray!=0: gather state1[F(i)] with F(i)=i^(i>>1)  (the CNOT-ladder fold).
// ---------------------------------------------------------------------------
__global__ void k_ulow(const float* __restrict__ srcRe, const float* __restrict__ srcIm,
                       float* __restrict__ dstRe, float* __restrict__ dstIm,
                       const float* __restrict__ cp, int layer, int gray)
{
    extern __shared__ float smem[];
    float* sre = smem;
    float* sim = smem + PTILE;
    __shared__ float Msh[16][17];
    __shared__ float rc[4], rs[4];

    int tid = threadIdx.x;
    unsigned blockBase = (unsigned)blockIdx.x << LOWB;
    unsigned grayBase  = blockBase ^ (blockBase >> 1);

    for (int i = 0; i < PER_THREAD; ++i) {
        unsigned local = (unsigned)i * NTHREADS + tid;
        unsigned src = gray ? (grayBase ^ (local ^ (local >> 1)))
                            : (blockBase | local);
        sre[phys(local)] = srcRe[src];
        sim[phys(local)] = srcIm[src];
    }
    __syncthreads();

    build_M(Msh, rc, rs, cp, layer, 0, 15, tid);
    apply_group<0>(sre, sim, Msh, tid);
    __syncthreads();
    build_M(Msh, rc, rs, cp, layer, 4, 15, tid);
    apply_group<4>(sre, sim, Msh, tid);
    __syncthreads();
    build_M(Msh, rc, rs, cp, layer, 8, 15, tid);
    apply_group<8>(sre, sim, Msh, tid);
    __syncthreads();
    // partial group: identity on bits 10-11 (done above), rotate bits 12-13
    build_M(Msh, rc, rs, cp, layer, 10, 12, tid);
    apply_group<10>(sre, sim, Msh, tid);
    __syncthreads();

    for (int i = 0; i < PER_THREAD; ++i) {
        unsigned local = (unsigned)i * NTHREADS + tid;
        unsigned d = blockBase | local;
        dstRe[d] = sre[phys(local)];
        dstIm[d] = sim[phys(local)];
    }
}

// ---------------------------------------------------------------------------
// K4/K6: high pass. Tile = all 2^8 values of bits 14..21  x  2^6 contiguous
// low bits (mid bits 6..13 fixed per block). Local index = (h<<6)|low6, so
// rotations on global bits 14..21 are WMMA stages at local g = 6 and 10.
// prob!=0: emit |amp|^2 * invn2 scattered through the inverse Gray map
// (prefix-XOR) -> folds the final CNOT ladder into the output permutation.
// ---------------------------------------------------------------------------
__global__ void k_uhigh(const float* __restrict__ srcRe, const float* __restrict__ srcIm,
                        float* __restrict__ dstRe, float* __restrict__ dstIm,
                        float* __restrict__ out, const float* __restrict__ invn2p,
                        const float* __restrict__ cp, int layer, int prob)
{
    extern __shared__ float smem[];
    float* sre = smem;
    float* sim = smem + PTILE;
    __shared__ float Msh[16][17];
    __shared__ float rc[4], rs[4];

    int tid = threadIdx.x;
    unsigned mid = (unsigned)blockIdx.x;          // bits 6..13 of global index

    for (int i = 0; i < PER_THREAD; ++i) {
        unsigned local = (unsigned)i * NTHREADS + tid;
        unsigned low6 = local & 63u;
        unsigned h    = local >> CONT;
        unsigned gidx = (h << LOWB) | (mid << CONT) | low6;
        sre[phys(local)] = srcRe[gidx];
        sim[phys(local)] = srcIm[gidx];
    }
    __syncthreads();

    build_M(Msh, rc, rs, cp, layer, 14, 15, tid);   // global bits 14..17
    apply_group<6>(sre, sim, Msh, tid);
    __syncthreads();
    build_M(Msh, rc, rs, cp, layer, 18, 15, tid);   // global bits 18..21
    apply_group<10>(sre, sim, Msh, tid);
    __syncthreads();

    float inv = invn2p[0];
    for (int i = 0; i < PER_THREAD; ++i) {
        unsigned local = (unsigned)i * NTHREADS + tid;
        unsigned low6 = local & 63u;
        unsigned h    = local >> CONT;
        unsigned gidx = (h << LOWB) | (mid << CONT) | low6;
        if (prob) {
            float r = sre[phys(local)], m = sim[phys(local)];
            unsigned x = gidx;                 // inverse Gray: prefix XOR
            x ^= x >> 1; x ^= x >> 2; x ^= x >> 4; x ^= x >> 8; x ^= x >> 16;
            out[x] = (r * r + m * m) * inv;
        } else {
            dstRe[gidx] = sre[phys(local)];
            dstIm[gidx] = sim[phys(local)];
        }
    }
}

// ---------------------------------------------------------------------------
extern "C" void kernel_launch(void* const* d_in, const int* in_sizes, int n_in,
                              void* d_out, int out_size, void* d_ws, size_t ws_size,
                              hipStream_t stream)
{
    const float* alpha = (const float*)d_in[0];
    const float* iaw   = (const float*)d_in[1];
    const float* w1    = (const float*)d_in[2];
    const float* b1    = (const float*)d_in[3];
    const float* w2    = (const float*)d_in[4];
    const float* b2    = (const float*)d_in[5];
    const float* w3    = (const float*)d_in[6];
    const float* b3    = (const float*)d_in[7];
    const float* cp    = (const float*)d_in[8];
    float* out = (float*)d_out;

    float* ws  = (float*)d_ws;           // needs ~64MB + 8.5KB
    float* Are = ws;
    float* Aim = ws + (size_t)DIM;
    float* Bre = ws + 2 * (size_t)DIM;
    float* Bim = ws + 3 * (size_t)DIM;
    float* invn2    = ws + 4 * (size_t)DIM;
    float* partials = ws + 4 * (size_t)DIM + 64;

    size_t shmem = 2u * PTILE * sizeof(float);   // 136 KB (WGP has 320 KB LDS)

    k_build<<<dim3(2048), dim3(NTHREADS), 0, stream>>>(
        alpha, iaw, w1, b1, w2, b2, w3, b3, Are, Aim, partials);
    k_norm<<<dim3(1), dim3(NTHREADS), 0, stream>>>(partials, invn2);

    // layer 1
    k_ulow <<<dim3(DIM / TILE), dim3(NTHREADS), shmem, stream>>>(
        Are, Aim, Are, Aim, cp, 0, /*gray=*/0);
    k_uhigh<<<dim3(DIM / TILE), dim3(NTHREADS), shmem, stream>>>(
        Are, Aim, Are, Aim, out, invn2, cp, 0, /*prob=*/0);
    // first CNOT ladder folded into the gather below; layer 2
    k_ulow <<<dim3(DIM / TILE), dim3(NTHREADS), shmem, stream>>>(
        Are, Aim, Bre, Bim, cp, 1, /*gray=*/1);
    // second CNOT ladder folded into the inverse-Gray output scatter
    k_uhigh<<<dim3(DIM / TILE), dim3(NTHREADS), shmem, stream>>>(
        Bre, Bim, Bre, Bim, out, invn2, cp, 1, /*prob=*/1);
}